// GCN_17446157156485
// MI455X (gfx1250) — compile-verified
//
#include <hip/hip_runtime.h>
#include <hip/hip_bf16.h>

#define NN   100000     // nodes
#define NE   3200000    // edges
#define FIN  2000       // input features
#define HD   16         // hidden dim
#define CD   3          // classes
#define NTILES (NN / 16)   // 6250 row tiles of 16
#define SCAN_T 1024

typedef __attribute__((ext_vector_type(2))) float v2f;
typedef __attribute__((ext_vector_type(8))) float v8f;

// ---------------- CSC build: histogram -> scan -> counting-sort scatter ----------------

__global__ void k_zero_counts(int* __restrict__ count) {
  int i = blockIdx.x * blockDim.x + threadIdx.x;
  if (i < NN) count[i] = 0;
}

__global__ void k_hist(const int* __restrict__ dst, int* __restrict__ count) {
  int e = blockIdx.x * blockDim.x + threadIdx.x;
  if (e < NE) atomicAdd(&count[dst[e]], 1);
}

// dinv[i] = rsqrt(in_degree + 1)  (self-loop)
__global__ void k_dinv(const int* __restrict__ count, float* __restrict__ dinv) {
  int i = blockIdx.x * blockDim.x + threadIdx.x;
  if (i < NN) dinv[i] = rsqrtf((float)(count[i] + 1));
}

// single-workgroup exclusive scan: count[0..NN) -> rowptr[0..NN]
__global__ void __launch_bounds__(SCAN_T)
k_scan(const int* __restrict__ count, int* __restrict__ rowptr) {
  __shared__ int lds[SCAN_T];
  const int t = threadIdx.x;
  const int chunk = (NN + SCAN_T - 1) / SCAN_T;          // 98
  const int beg = t * chunk;
  const int end = (beg + chunk < NN) ? beg + chunk : NN;
  int s = 0;
  for (int i = beg; i < end && i < NN; ++i) s += count[i];
  lds[t] = s;
  __syncthreads();
  // Hillis-Steele inclusive scan over 1024 partial sums
  for (int off = 1; off < SCAN_T; off <<= 1) {
    int v = (t >= off) ? lds[t - off] : 0;
    __syncthreads();
    lds[t] += v;
    __syncthreads();
  }
  int prefix = (t == 0) ? 0 : lds[t - 1];                // exclusive
  for (int i = beg; i < end && i < NN; ++i) {
    rowptr[i] = prefix;
    prefix += count[i];
  }
  if (t == SCAN_T - 1) rowptr[NN] = lds[SCAN_T - 1];     // == NE
}

__global__ void k_copy_cursor(const int* __restrict__ rowptr, int* __restrict__ cursor) {
  int i = blockIdx.x * blockDim.x + threadIdx.x;
  if (i < NN) cursor[i] = rowptr[i];
}

// counting sort: col[] holds source node of every edge, grouped by destination
__global__ void k_scatter(const int* __restrict__ src, const int* __restrict__ dst,
                          int* __restrict__ cursor, int* __restrict__ col) {
  int e = blockIdx.x * blockDim.x + threadIdx.x;
  if (e < NE) {
    int pos = atomicAdd(&cursor[dst[e]], 1);
    col[pos] = src[e];
  }
}

// ---------------- layer 1 GEMM: h1 = x @ W1 via V_WMMA_F32_16X16X4_F32 ----------------
// One wave computes one 16x16 output tile. Wave32 layouts per CDNA5 ISA 7.12.2:
// A (16x4 f32): lane&15 = row M; VGPR pair holds K = k0 + 2*(lane>=16) + {0,1}.
// B (4x16 f32): lane&15 = col N; same K split across lane halves.
// C/D (16x16 f32, 8 VGPRs): VGPR r -> M = r + 8*(lane>=16), N = lane&15.

__global__ void k_gemm1(const float* __restrict__ x,
                        const float* __restrict__ W1,
                        float* __restrict__ h1) {
  const int lane = threadIdx.x & 31;
  const int tile = blockIdx.x * (blockDim.x >> 5) + (threadIdx.x >> 5);
  if (tile >= NTILES) return;            // uniform per wave -> EXEC all-1 for active waves

  const int half = lane >> 4;            // 0 or 1
  const int m    = lane & 15;            // row (A) / col (B/C/D)

  const float* __restrict__ arow = x + (size_t)(tile * 16 + m) * FIN + 2 * half;
  const float* __restrict__ bcol = W1 + m;   // column m of W1, row stride HD

  v8f acc = {0.f, 0.f, 0.f, 0.f, 0.f, 0.f, 0.f, 0.f};

  for (int k0 = 0; k0 < FIN; k0 += 4) {
    v2f a = *(const v2f*)(arow + k0);    // 8B-aligned (row stride 8000B)
    const int kb = k0 + 2 * half;
    v2f b;
    b.x = bcol[(size_t)(kb + 0) * HD];
    b.y = bcol[(size_t)(kb + 1) * HD];
    acc = __builtin_amdgcn_wmma_f32_16x16x4_f32(
        /*neg_a=*/false, a, /*neg_b=*/false, b,
        /*c_mod=*/(short)0, acc, /*reuse_a=*/false, /*reuse_b=*/false);
  }

  float* __restrict__ orow = h1 + (size_t)(tile * 16 + 8 * half) * HD + m;
#pragma unroll
  for (int r = 0; r < 8; ++r) orow[(size_t)r * HD] = acc[r];
}

// ---------------- layer 1 aggregation: gather-sum (wave per node) + bias + relu -------
// Lanes: f = lane&15 (feature), h = lane>>4 (neighbor slice); each slice walks every
// 2nd neighbor; per-neighbor the 16 feature lanes read a contiguous 64B row of h1.

__global__ void k_gather1(const float* __restrict__ h1, const float* __restrict__ dinv,
                          const int* __restrict__ rowptr, const int* __restrict__ col,
                          const float* __restrict__ b1, float* __restrict__ hrelu) {
  const int node = blockIdx.x * (blockDim.x >> 5) + (threadIdx.x >> 5);
  if (node >= NN) return;
  const int lane = threadIdx.x & 31;
  const int f = lane & 15;
  const int h = lane >> 4;

  const int beg = rowptr[node];
  const int end = rowptr[node + 1];

  float acc = 0.f;
  for (int j = beg + h; j < end; j += 2) {
    int s = col[j];
    acc += dinv[s] * h1[(size_t)s * HD + f];
  }
  acc += __shfl_xor(acc, 16, 32);        // combine the two neighbor slices (wave32)

  const float di = dinv[node];
  float v = di * (acc + di * h1[(size_t)node * HD + f]) + b1[f];
  if (h == 0) hrelu[(size_t)node * HD + f] = v > 0.f ? v : 0.f;
}

// ---------------- layer 2 (16 -> 3) ----------------

__global__ void k_lin2(const float* __restrict__ hr, const float* __restrict__ W2,
                       float* __restrict__ h2) {
  int i = blockIdx.x * blockDim.x + threadIdx.x;
  if (i >= NN) return;
  float a0 = 0.f, a1 = 0.f, a2 = 0.f;
#pragma unroll
  for (int f = 0; f < HD; ++f) {
    float v = hr[(size_t)i * HD + f];
    a0 += v * W2[f * CD + 0];
    a1 += v * W2[f * CD + 1];
    a2 += v * W2[f * CD + 2];
  }
  h2[(size_t)i * CD + 0] = a0;
  h2[(size_t)i * CD + 1] = a1;
  h2[(size_t)i * CD + 2] = a2;
}

// gather-sum + bias + log-softmax fused, thread per node, writes final output
__global__ void k_gather2(const float* __restrict__ h2, const float* __restrict__ dinv,
                          const int* __restrict__ rowptr, const int* __restrict__ col,
                          const float* __restrict__ b2, float* __restrict__ out) {
  int i = blockIdx.x * blockDim.x + threadIdx.x;
  if (i >= NN) return;
  float a0 = 0.f, a1 = 0.f, a2 = 0.f;
  const int beg = rowptr[i];
  const int end = rowptr[i + 1];
  for (int j = beg; j < end; ++j) {
    int s = col[j];
    float w = dinv[s];
    a0 += w * h2[(size_t)s * CD + 0];
    a1 += w * h2[(size_t)s * CD + 1];
    a2 += w * h2[(size_t)s * CD + 2];
  }
  const float di = dinv[i];
  float l0 = di * (a0 + di * h2[(size_t)i * CD + 0]) + b2[0];
  float l1 = di * (a1 + di * h2[(size_t)i * CD + 1]) + b2[1];
  float l2 = di * (a2 + di * h2[(size_t)i * CD + 2]) + b2[2];
  float m = fmaxf(l0, fmaxf(l1, l2));
  float lse = m + logf(expf(l0 - m) + expf(l1 - m) + expf(l2 - m));
  out[(size_t)i * CD + 0] = l0 - lse;
  out[(size_t)i * CD + 1] = l1 - lse;
  out[(size_t)i * CD + 2] = l2 - lse;
}

// ---------------- launcher ----------------

extern "C" void kernel_launch(void* const* d_in, const int* in_sizes, int n_in,
                              void* d_out, int out_size, void* d_ws, size_t ws_size,
                              hipStream_t stream) {
  const float* x   = (const float*)d_in[0];
  const int*   ei  = (const int*)d_in[1];        // [2, E]
  const float* W1  = (const float*)d_in[2];
  const float* b1  = (const float*)d_in[3];
  const float* W2  = (const float*)d_in[4];
  const float* b2  = (const float*)d_in[5];
  const int* src = ei;
  const int* dst = ei + NE;
  float* out = (float*)d_out;

  // workspace carve-out (128B-aligned offsets). count[] is reused as the scatter cursor.
  char* w = (char*)d_ws;
  int*   count  = (int*)(w + 0);                 // NN ints (later: cursor)
  float* dinv   = (float*)(w + 400128);          // NN floats
  int*   rowptr = (int*)(w + 800256);            // NN+1 ints
  int*   col    = (int*)(w + 1200512);           // NE ints (CSC columns = src ids)
  float* h1     = (float*)(w + 14000512);        // NN*HD floats
  float* hrelu  = (float*)(w + 20400512);        // NN*HD floats
  float* h2     = (float*)(w + 26800512);        // NN*CD floats  (end ~28.0 MB)

  const int B = 256;
  const int gN  = (NN + B - 1) / B;
  const int gE  = (NE + B - 1) / B;

  // CSC build + norms
  k_zero_counts<<<gN, B, 0, stream>>>(count);
  k_hist       <<<gE, B, 0, stream>>>(dst, count);
  k_dinv       <<<gN, B, 0, stream>>>(count, dinv);
  k_scan       <<<1, SCAN_T, 0, stream>>>(count, rowptr);
  k_copy_cursor<<<gN, B, 0, stream>>>(rowptr, count);   // count becomes cursor
  k_scatter    <<<gE, B, 0, stream>>>(src, dst, count, col);

  // layer 1: WMMA GEMM then gather-aggregate (+bias+relu fused)
  {
    int wavesPerBlock = B / 32;
    int blocks = (NTILES + wavesPerBlock - 1) / wavesPerBlock;
    k_gemm1<<<blocks, B, 0, stream>>>(x, W1, h1);
  }
  {
    int wavesPerBlock = B / 32;
    int blocks = (NN + wavesPerBlock - 1) / wavesPerBlock;   // 12500
    k_gather1<<<blocks, B, 0, stream>>>(h1, dinv, rowptr, col, b1, hrelu);
  }

  // layer 2: tiny GEMM, then gather + bias + log-softmax fused into output
  k_lin2   <<<gN, B, 0, stream>>>(hrelu, W2, h2);
  k_gather2<<<gN, B, 0, stream>>>(h2, dinv, rowptr, col, b2, out);
}